// HyperButterflyLayer_54202487275783
// MI455X (gfx1250) — compile-verified
//
#include <hip/hip_runtime.h>
#include <math.h>

#define DIM  4096
#define FEAT 4096
#define HALF 2048
#define NUM_LAYERS 12
#define EPSV 1e-8f

typedef __attribute__((ext_vector_type(2))) float v2f;
typedef __attribute__((ext_vector_type(4))) float f4;
typedef __attribute__((ext_vector_type(8))) float v8f;

// Workspace layout (float offsets)
#define WS_SIN  0                 // 4096  per-row input scale atanh(n)/n
#define WS_CA   4096              // 2048  combined pair coeff A
#define WS_CB   6144              // 2048  combined pair coeff B
#define WS_PART 8192              // 4096*4 partial sumsq of output rows
#define WS_SOUT (8192 + 16384)    // 4096  per-row output scale tanh(n)/n

// LDS tile row pitch in floats: 260 mod 64 == 4 -> 16 rows hit distinct bank
// groups for the strided WMMA-A reads; 260*4 B keeps rows 16B-aligned for b128.
#define TSTRIDE 260
#define TILEC   256               // columns staged per tile
#define NTILE   (FEAT / TILEC)    // 16 tiles of 16 rows x 256 cols

// ---------------------------------------------------------------------------
// K1: per-row sum of squares of X via V_WMMA_F32_16X16X4_F32, fed from LDS
// tiles staged with GLOBAL_LOAD_ASYNC_TO_LDS_B128 (ASYNCcnt, double-buffered).
// A[m][k] = x^2 (16x4 layout: lanes 0-15 -> K=0,1 ; lanes 16-31 -> K=2,3 ;
// M = lane&15), B = all-ones so D[m][n] = C + sum_k A[m][k] for every n.
// One 16-row group per block; 8 waves split each tile's 256 columns.
// ---------------------------------------------------------------------------
__global__ void k1_rownorm_wmma(const float* __restrict__ X, float* __restrict__ WS) {
    __shared__ float tile[2][16 * TSTRIDE];
    __shared__ float part[8][16];

    const int tid  = threadIdx.x;
    const int lane = tid & 31;
    const int wave = tid >> 5;
    const int r0   = blockIdx.x * 16;

    // Async-copy indexing: 1024 float4 per tile; thread tid moves 4 of them.
    // idx = tid + 256*j -> row16 = idx>>6, col4 = idx&63 (coalesced per row).
    unsigned ldsoff[4];
    const float* gptr[4];
    #pragma unroll
    for (int j = 0; j < 4; ++j) {
        const int idx   = tid + 256 * j;
        const int row16 = idx >> 6;
        const int col4  = idx & 63;
        ldsoff[j] = (unsigned)(row16 * TSTRIDE + col4 * 4) * 4u;
        gptr[j]   = X + (size_t)(r0 + row16) * FEAT + col4 * 4;
    }
    // Low 32 bits of a generic LDS pointer == wave-relative LDS byte offset.
    const unsigned ldsbase = (unsigned)(uintptr_t)(&tile[0][0]);
    const unsigned bufstep = (unsigned)(16 * TSTRIDE * 4);

    #pragma unroll
    for (int j = 0; j < 4; ++j)
        asm volatile("global_load_async_to_lds_b128 %0, %1, off"
                     :: "v"(ldsbase + ldsoff[j]), "v"(gptr[j]) : "memory");

    const int row  = lane & 15;
    const int koff = (lane >> 4) << 1;          // K-pair select per lane half
    const int tbase = row * TSTRIDE + wave * 32 + koff;

    v8f c = {0.f,0.f,0.f,0.f,0.f,0.f,0.f,0.f};
    v2f b; b.x = 1.0f; b.y = 1.0f;              // B = ones (layout-independent)

    for (int t = 0; t < NTILE; ++t) {
        if (t + 1 < NTILE) {
            const unsigned dstb = ldsbase + (unsigned)((t + 1) & 1) * bufstep;
            #pragma unroll
            for (int j = 0; j < 4; ++j)
                asm volatile("global_load_async_to_lds_b128 %0, %1, off"
                             :: "v"(dstb + ldsoff[j]),
                                "v"(gptr[j] + (size_t)(t + 1) * TILEC) : "memory");
            asm volatile("s_wait_asynccnt 0x4" ::: "memory");  // tile t landed
        } else {
            asm volatile("s_wait_asynccnt 0x0" ::: "memory");
        }
        __syncthreads();                         // everyone's copies visible

        const int tb = t & 1;
        #pragma unroll
        for (int cc = 0; cc < 32; cc += 4) {
            const float x0 = tile[tb][tbase + cc];
            const float x1 = tile[tb][tbase + cc + 1];
            v2f a; a.x = x0 * x0; a.y = x1 * x1;
            c = __builtin_amdgcn_wmma_f32_16x16x4_f32(
                    /*neg_a=*/false, a, /*neg_b=*/false, b,
                    /*c_mod=*/(short)0, c, /*reuse_a=*/false, /*reuse_b=*/false);
        }
        __syncthreads();                         // safe to overwrite buffer
    }

    // D layout: VGPR g, lanes 0-15 -> row r0+g ; lanes 16-31 -> row r0+8+g.
    if (lane == 0)  { for (int g = 0; g < 8; ++g) part[wave][g]     = c[g]; }
    if (lane == 16) { for (int g = 0; g < 8; ++g) part[wave][8 + g] = c[g]; }
    __syncthreads();
    if (tid < 16) {
        float s = 0.f;
        for (int w = 0; w < 8; ++w) s += part[w][tid];
        const float n = sqrtf(s);
        WS[WS_SIN + r0 + tid] = (n > 1e-20f) ? (atanhf(n) / n) : 1.0f;
    }
}

// ---------------------------------------------------------------------------
// Kc: collapse layers 1..11 into one (A,B) per pair via complex products.
// Each layer acts as multiplication by (a - i b); identity past k = DIM>>l.
// ---------------------------------------------------------------------------
__global__ void k_combine(const float* __restrict__ P, float* __restrict__ WS) {
    const int p = blockIdx.x * blockDim.x + threadIdx.x;
    if (p >= HALF) return;
    const int OFF[NUM_LAYERS] = {0, 8192, 12288, 14336, 15360, 15872,
                                 16128, 16256, 16320, 16352, 16368, 16376};
    float cr = 1.f, ci = 0.f;
    #pragma unroll
    for (int l = 1; l < NUM_LAYERS; ++l) {
        const int k = DIM >> l;
        float a = 1.f, bb = 0.f;
        if (p < k) { a = P[OFF[l] + 2 * p]; bb = P[OFF[l] + 2 * p + 1]; }
        const float nr = cr * a + ci * bb;       // (cr + i ci) * (a - i bb)
        const float ni = ci * a - cr * bb;
        cr = nr; ci = ni;
    }
    WS[WS_CA + p] = cr;    // y0 =  A u0 + B u1
    WS[WS_CB + p] = -ci;   // y1 = -B u0 + A u1
}

// ---------------------------------------------------------------------------
// K2: fused input scale + layer0 (rows i,i+1) + combined pair rotation.
// One pair per blockIdx.y, 1024-column tile per blockIdx.x, float4 per thread.
// Writes v to OUT and deterministic per-(row,tile) sumsq partials to WS.
// ---------------------------------------------------------------------------
__global__ void k2_butterfly(const float* __restrict__ X, const float* __restrict__ P,
                             float* __restrict__ WS, float* __restrict__ OUT) {
    const int tid = threadIdx.x;
    const int col = (blockIdx.x * 256 + tid) * 4;
    const int p   = blockIdx.y;
    const int r0 = 2 * p, r1 = r0 + 1, r2 = r0 + 2;

    const float s0 = WS[WS_SIN + r0];
    const float s1 = WS[WS_SIN + r1];
    const f4 x0 = *(const f4*)(X + (size_t)r0 * FEAT + col);
    const f4 x1 = *(const f4*)(X + (size_t)r1 * FEAT + col);
    const f4 u0 = s0 * x0;
    const f4 u1 = s1 * x1;

    const float a00 = P[2 * r0], b00 = P[2 * r0 + 1];
    f4 t0 = a00 * u0 + b00 * u1;
    f4 t1;
    if (r1 < DIM - 1) {                         // uniform per block
        const float s2 = WS[WS_SIN + r2];
        const f4 x2 = *(const f4*)(X + (size_t)r2 * FEAT + col);
        const f4 u2 = s2 * x2;
        const float a01 = P[2 * r1], b01 = P[2 * r1 + 1];
        t1 = a01 * u1 + b01 * u2;
    } else {                                    // last row: -b[-1]*u[-2] + a[-1]*u[-1]
        t1 = -b00 * u0 + a00 * u1;
    }

    const float A = WS[WS_CA + p], B = WS[WS_CB + p];
    const f4 v0 =  A * t0 + B * t1;
    const f4 v1 = -B * t0 + A * t1;
    *(f4*)(OUT + (size_t)r0 * FEAT + col) = v0;
    *(f4*)(OUT + (size_t)r1 * FEAT + col) = v1;

    // deterministic block-level sumsq partials (no atomics -> replay-stable)
    const f4 q0 = v0 * v0, q1 = v1 * v1;
    float w0 = q0.x + q0.y + q0.z + q0.w;
    float w1 = q1.x + q1.y + q1.z + q1.w;
    for (int off = 16; off > 0; off >>= 1) {
        w0 += __shfl_down(w0, off);
        w1 += __shfl_down(w1, off);
    }
    __shared__ float a0s[8], a1s[8];
    const int lane = tid & 31, wave = tid >> 5;
    if (lane == 0) { a0s[wave] = w0; a1s[wave] = w1; }
    __syncthreads();
    if (tid == 0) {
        float q0t = 0.f, q1t = 0.f;
        for (int w = 0; w < 8; ++w) { q0t += a0s[w]; q1t += a1s[w]; }
        WS[WS_PART + r0 * 4 + blockIdx.x] = q0t;
        WS[WS_PART + r1 * 4 + blockIdx.x] = q1t;
    }
}

// ---------------------------------------------------------------------------
// K3: finish output row norms -> tanh(vn)/vn scales.
// ---------------------------------------------------------------------------
__global__ void k3_outscale(float* __restrict__ WS) {
    const int row = blockIdx.x * blockDim.x + threadIdx.x;
    if (row >= DIM) return;
    float s = 0.f;
    for (int t = 0; t < 4; ++t) s += WS[WS_PART + row * 4 + t];
    const float vn = fmaxf(sqrtf(s), EPSV);
    WS[WS_SOUT + row] = tanhf(vn) / vn;
}

// ---------------------------------------------------------------------------
// K4: in-place per-row rescale of OUT, float4 vectorized.
// ---------------------------------------------------------------------------
__global__ void k4_scale(const float* __restrict__ WS, float* __restrict__ OUT) {
    const int col = (blockIdx.x * 256 + threadIdx.x) * 4;
    const int row = blockIdx.y;
    const float sc = WS[WS_SOUT + row];
    __builtin_prefetch(OUT + (size_t)row * FEAT + col + 1024, 0, 1);
    f4* o = (f4*)(OUT + (size_t)row * FEAT + col);
    *o = *o * sc;
}

extern "C" void kernel_launch(void* const* d_in, const int* in_sizes, int n_in,
                              void* d_out, int out_size, void* d_ws, size_t ws_size,
                              hipStream_t stream) {
    const float* X = (const float*)d_in[0];   // (4096, 4096)
    const float* P = (const float*)d_in[1];   // (16380,)
    float* OUT = (float*)d_out;
    float* WS  = (float*)d_ws;

    k1_rownorm_wmma<<<dim3(DIM / 16), dim3(256), 0, stream>>>(X, WS);
    k_combine     <<<dim3(HALF / 256), dim3(256), 0, stream>>>(P, WS);
    k2_butterfly  <<<dim3(FEAT / 1024, HALF), dim3(256), 0, stream>>>(X, P, WS, OUT);
    k3_outscale   <<<dim3(DIM / 256), dim3(256), 0, stream>>>(WS);
    k4_scale      <<<dim3(FEAT / 1024, DIM), dim3(256), 0, stream>>>(WS, OUT);
}